// CausalSelfAttentionHead_6047313953342
// MI455X (gfx1250) — compile-verified
//
#include <hip/hip_runtime.h>
#include <math.h>

#define B_  2
#define T_  1024
#define E_  256
#define D_  64

typedef float v2f __attribute__((ext_vector_type(2)));
typedef float v8f __attribute__((ext_vector_type(8)));
typedef int   v4i __attribute__((ext_vector_type(4)));

#if defined(__has_builtin)
#  if __has_builtin(__builtin_amdgcn_global_load_async_to_lds_b128)
#    define HAVE_ASYNC_LDS 1
#  endif
#endif
#ifndef HAVE_ASYNC_LDS
#  define HAVE_ASYNC_LDS 0
#endif

#if HAVE_ASYNC_LDS
typedef __attribute__((address_space(1))) v4i gv4i;   // global int4
typedef __attribute__((address_space(3))) v4i lv4i;   // LDS int4
#endif

__device__ __forceinline__ v8f wmma4(v2f a, v2f b, v8f c) {
  // D = A(16x4 f32) * B(4x16 f32) + C(16x16 f32)
  return __builtin_amdgcn_wmma_f32_16x16x4_f32(false, a, false, b, (short)0, c, false, false);
}

// ---------------------------------------------------------------------------
// Kernel 1: per 16-row tile of flattened [B*T] rows, compute
//   q = relu(x@Wq+bq), k = relu(x@Wk+bk), v = x@Wv+bv          (stage 1, WMMA)
//   qpb = q@W1[:64] + b1,  kpo = k@W1[64:]                      (stage 2, WMMA)
// q/k bounce through LDS (padded stride 68) to re-fragment for stage 2.
// ---------------------------------------------------------------------------
__global__ __launch_bounds__(32) void proj_kernel(
    const float* __restrict__ x,
    const float* __restrict__ Wq, const float* __restrict__ bq,
    const float* __restrict__ Wk, const float* __restrict__ bk,
    const float* __restrict__ Wv, const float* __restrict__ bv,
    const float* __restrict__ W1, const float* __restrict__ b1,
    float* __restrict__ qpb, float* __restrict__ kpo, float* __restrict__ vvo)
{
  __shared__ float qs[16 * 68];
  __shared__ float ks[16 * 68];

  const int lane = threadIdx.x;
  const int c = lane & 15;     // M (A) / N (B,C) index within tile
  const int h = lane >> 4;     // K-half selector for A/B fragments
  const int m0 = blockIdx.x * 16;

  const float* xrow = x + (size_t)(m0 + c) * E_;

  const v8f zero8 = {0.f,0.f,0.f,0.f,0.f,0.f,0.f,0.f};
  v8f qa[4], ka[4], va[4];
#pragma unroll
  for (int nt = 0; nt < 4; ++nt) { qa[nt] = zero8; ka[nt] = zero8; va[nt] = zero8; }

  // ---- stage 1: K = 256 reduction, 4 output col-tiles of 16 ----
  for (int k0 = 0; k0 < E_; k0 += 4) {
    const int kk = k0 + 2 * h;
    const v2f a = *(const v2f*)(xrow + kk);   // A frag: x[m0+c][kk], x[m0+c][kk+1]
#pragma unroll
    for (int nt = 0; nt < 4; ++nt) {
      const int n = nt * 16 + c;
      v2f bw;
      bw.x = Wq[(size_t)kk * D_ + n]; bw.y = Wq[(size_t)(kk + 1) * D_ + n];
      qa[nt] = wmma4(a, bw, qa[nt]);
      bw.x = Wk[(size_t)kk * D_ + n]; bw.y = Wk[(size_t)(kk + 1) * D_ + n];
      ka[nt] = wmma4(a, bw, ka[nt]);
      bw.x = Wv[(size_t)kk * D_ + n]; bw.y = Wv[(size_t)(kk + 1) * D_ + n];
      va[nt] = wmma4(a, bw, va[nt]);
    }
  }

  // ---- stage 1 epilogue: bias (+relu for q,k), q/k->LDS, v->global ----
#pragma unroll
  for (int nt = 0; nt < 4; ++nt) {
    const int n = nt * 16 + c;
    const float bqv = bq[n], bkv = bk[n], bvv = bv[n];
#pragma unroll
    for (int r = 0; r < 8; ++r) {
      const int m = r + 8 * h;                 // C-layout row
      qs[m * 68 + n] = fmaxf(qa[nt][r] + bqv, 0.f);
      ks[m * 68 + n] = fmaxf(ka[nt][r] + bkv, 0.f);
      vvo[(size_t)(m0 + m) * D_ + n] = va[nt][r] + bvv;
    }
  }
  __syncthreads();

  // ---- stage 2: K = 64 reduction against W1 halves ----
  v8f qpa[4], kpa[4];
#pragma unroll
  for (int nt = 0; nt < 4; ++nt) { qpa[nt] = zero8; kpa[nt] = zero8; }

  for (int k0 = 0; k0 < D_; k0 += 4) {
    const int kk = k0 + 2 * h;
    const v2f aq = *(const v2f*)(qs + c * 68 + kk);
    const v2f ak = *(const v2f*)(ks + c * 68 + kk);
#pragma unroll
    for (int nt = 0; nt < 4; ++nt) {
      const int n = nt * 16 + c;
      v2f bw;
      bw.x = W1[(size_t)kk * D_ + n];            bw.y = W1[(size_t)(kk + 1) * D_ + n];
      qpa[nt] = wmma4(aq, bw, qpa[nt]);
      bw.x = W1[(size_t)(64 + kk) * D_ + n];     bw.y = W1[(size_t)(64 + kk + 1) * D_ + n];
      kpa[nt] = wmma4(ak, bw, kpa[nt]);
    }
  }

#pragma unroll
  for (int nt = 0; nt < 4; ++nt) {
    const int n = nt * 16 + c;
    const float b1v = b1[n];                     // fold b1 into qp once
#pragma unroll
    for (int r = 0; r < 8; ++r) {
      const int m = r + 8 * h;
      qpb[(size_t)(m0 + m) * D_ + n] = qpa[nt][r] + b1v;
      kpo[(size_t)(m0 + m) * D_ + n] = kpa[nt][r];
    }
  }
}

// ---------------------------------------------------------------------------
// Kernel 2: fused additive-attention scores + causal online softmax + P@V.
// One wave per 32 query rows; lane = query row. qp row + O row in VGPRs,
// W2 in SGPRs, K/V tiles staged in LDS via async-to-LDS (ASYNCcnt) when
// available, read with uniform-address ds_load_b128 broadcast.
// (b2 dropped: uniform pre-softmax shift cancels in softmax.)
// ---------------------------------------------------------------------------
__global__ __launch_bounds__(32) void attn_kernel(
    const float* __restrict__ qpb, const float* __restrict__ kpo,
    const float* __restrict__ vvo, const float* __restrict__ W2,
    float* __restrict__ out)
{
  __shared__ float4 ksh[16 * 16];   // 16 rows x 64 floats
  __shared__ float4 vsh[16 * 16];

  const int lane = threadIdx.x;
  const int bb = blockIdx.x >> 5;               // T_/32 = 32 query tiles per batch
  const int i0 = (blockIdx.x & 31) * 32;
  const int i  = i0 + lane;
  const size_t rowq = (size_t)(bb * T_ + i);

  // W2 -> SGPRs (wave-uniform)
  float w2s[64];
#pragma unroll
  for (int d = 0; d < 64; ++d)
    w2s[d] = __int_as_float(__builtin_amdgcn_readfirstlane(__float_as_int(W2[d])));

  // this lane's qp row (b1 already folded in)
  float qpv[64];
  {
    const float4* qr = (const float4*)(qpb + rowq * D_);
#pragma unroll
    for (int q = 0; q < 16; ++q) {
      const float4 t = qr[q];
      qpv[4*q+0] = t.x; qpv[4*q+1] = t.y; qpv[4*q+2] = t.z; qpv[4*q+3] = t.w;
    }
  }

  float O[64];
#pragma unroll
  for (int d = 0; d < 64; ++d) O[d] = 0.f;
  float mrun = -INFINITY, ssum = 0.f;

  const int jr = lane >> 1, hf = lane & 1;      // cooperative tile-load mapping
  const int ntile = (i0 >> 4) + 2;              // causal: keys up to i0+31

  for (int kt = 0; kt < ntile; ++kt) {
    const int j0 = kt << 4;
    {
      const size_t base = ((size_t)(bb * T_ + j0 + jr) * D_ + hf * 32);
      const float4* kg = (const float4*)(kpo + base);
      const float4* vg = (const float4*)(vvo + base);
      float4* kd = ksh + (jr * 16 + hf * 8);
      float4* vd = vsh + (jr * 16 + hf * 8);
#if HAVE_ASYNC_LDS
      // WAR: previous tile's ds_loads must have drained before the async
      // engine overwrites LDS (async writes are unordered with DS ops).
      asm volatile("s_wait_dscnt 0" ::: "memory");
#pragma unroll
      for (int q = 0; q < 8; ++q)
        __builtin_amdgcn_global_load_async_to_lds_b128(
            (gv4i*)(kg + q), (lv4i*)(kd + q), 0, 0);
#pragma unroll
      for (int q = 0; q < 8; ++q)
        __builtin_amdgcn_global_load_async_to_lds_b128(
            (gv4i*)(vg + q), (lv4i*)(vd + q), 0, 0);
      asm volatile("s_wait_asynccnt 0" ::: "memory");
#else
#pragma unroll
      for (int q = 0; q < 8; ++q) kd[q] = kg[q];
#pragma unroll
      for (int q = 0; q < 8; ++q) vd[q] = vg[q];
#endif
    }
    if (kt + 1 < ntile) {                       // prefetch next K/V tile
      const size_t nb = ((size_t)(bb * T_ + j0 + 16 + jr) * D_ + hf * 32);
      __builtin_prefetch(kpo + nb, 0, 0);
      __builtin_prefetch(vvo + nb, 0, 0);
    }
    __syncthreads();

    const bool needmask = (j0 + 15 > i0);
    float s[16];
#pragma unroll
    for (int jj = 0; jj < 16; ++jj) {
      float acc = 0.f;
#pragma unroll
      for (int ch = 0; ch < 16; ++ch) {
        const float4 kv = ksh[jj * 16 + ch];    // uniform-address LDS broadcast
        acc = fmaf(w2s[4*ch+0], fmaxf(qpv[4*ch+0] + kv.x, 0.f), acc);
        acc = fmaf(w2s[4*ch+1], fmaxf(qpv[4*ch+1] + kv.y, 0.f), acc);
        acc = fmaf(w2s[4*ch+2], fmaxf(qpv[4*ch+2] + kv.z, 0.f), acc);
        acc = fmaf(w2s[4*ch+3], fmaxf(qpv[4*ch+3] + kv.w, 0.f), acc);
      }
      s[jj] = (needmask && (j0 + jj > i)) ? -INFINITY : acc;
    }

    // online softmax (per-lane; lane == query row, no cross-lane reduction)
    float tmax = s[0];
#pragma unroll
    for (int jj = 1; jj < 16; ++jj) tmax = fmaxf(tmax, s[jj]);
    const float mnew  = fmaxf(mrun, tmax);
    const float scale = __expf(mrun - mnew);
#pragma unroll
    for (int d = 0; d < 64; ++d) O[d] *= scale;

    float psum = 0.f;
#pragma unroll
    for (int jj = 0; jj < 16; ++jj) {
      const float p = __expf(s[jj] - mnew);
      psum += p;
#pragma unroll
      for (int ch = 0; ch < 16; ++ch) {
        const float4 vv = vsh[jj * 16 + ch];
        O[4*ch+0] = fmaf(p, vv.x, O[4*ch+0]);
        O[4*ch+1] = fmaf(p, vv.y, O[4*ch+1]);
        O[4*ch+2] = fmaf(p, vv.z, O[4*ch+2]);
        O[4*ch+3] = fmaf(p, vv.w, O[4*ch+3]);
      }
    }
    ssum = ssum * scale + psum;
    mrun = mnew;
    __syncthreads();
  }

  const float inv = 1.f / ssum;
  float4* orow = (float4*)(out + rowq * D_);
#pragma unroll
  for (int q = 0; q < 16; ++q) {
    float4 t;
    t.x = O[4*q+0]*inv; t.y = O[4*q+1]*inv; t.z = O[4*q+2]*inv; t.w = O[4*q+3]*inv;
    orow[q] = t;
  }
}

// ---------------------------------------------------------------------------
extern "C" void kernel_launch(void* const* d_in, const int* in_sizes, int n_in,
                              void* d_out, int out_size, void* d_ws, size_t ws_size,
                              hipStream_t stream) {
  (void)in_sizes; (void)n_in; (void)out_size; (void)ws_size;
  const float* x  = (const float*)d_in[0];
  const float* Wq = (const float*)d_in[1];
  const float* bq = (const float*)d_in[2];
  const float* Wk = (const float*)d_in[3];
  const float* bk = (const float*)d_in[4];
  const float* Wv = (const float*)d_in[5];
  const float* bv = (const float*)d_in[6];
  const float* W1 = (const float*)d_in[7];
  const float* b1 = (const float*)d_in[8];
  const float* W2 = (const float*)d_in[9];
  // d_in[10] (b2): uniform pre-softmax shift, mathematically a no-op.

  float* ws  = (float*)d_ws;
  float* qpb = ws;                                // [B*T,64]
  float* kpo = ws + (size_t)B_ * T_ * D_;         // [B*T,64]
  float* vvo = ws + 2 * (size_t)B_ * T_ * D_;     // [B*T,64]
  float* out = (float*)d_out;

  proj_kernel<<<(B_ * T_) / 16, 32, 0, stream>>>(x, Wq, bq, Wk, bk, Wv, bv,
                                                 W1, b1, qpb, kpo, vvo);
  attn_kernel<<<B_ * (T_ / 32), 32, 0, stream>>>(qpb, kpo, vvo, W2, out);
}